// MyModel_51677046505874
// MI455X (gfx1250) — compile-verified
//
#include <hip/hip_runtime.h>
#include <hip/hip_bf16.h>

// Problem constants (from reference)
#define N_NODES   100000
#define N_EDGES   3200000
#define F_IN      512
#define HIDDEN    16
#define N_CLASSES 32

typedef __attribute__((ext_vector_type(2))) float v2f;
typedef __attribute__((ext_vector_type(8))) float v8f;

// Deterministic hash producing 2 Bernoulli(0.5) bits for an even/odd element
// pair; dropout rate=0.5 -> keep scale 2.0.
__device__ __forceinline__ unsigned pair_hash(unsigned pair_idx, unsigned seed) {
    unsigned x = pair_idx ^ seed;
    x *= 0x9E3779B9u; x ^= x >> 16;
    x *= 0x85EBCA6Bu; x ^= x >> 13;
    return x;
}

__device__ __forceinline__ v2f dropout_pair(v2f v, unsigned elem_idx, unsigned seed) {
    const unsigned h = pair_hash(elem_idx >> 1, seed);
    v2f r;
    r.x = (h & 1u) ? 2.0f * v.x : 0.0f;
    r.y = (h & 2u) ? 2.0f * v.y : 0.0f;
    return r;
}

// ---------------------------------------------------------------------------
// Layer 1 GEMM: H1 = dropout(X) @ W1   [N,512]@[512,16] -> [N,16]
// One wave = one 16-row tile, f32 WMMA 16x16x4, 128 k-steps.
// X is streamed once with non-temporal hints (204.8 MB > 192 MB L2): keep it
// out of L2 so the SpMM working set (H matrices + edge lists) stays resident.
// ---------------------------------------------------------------------------
__global__ __launch_bounds__(256)
void gcn_gemm1_dropout_wmma(const float* __restrict__ X,
                            const float* __restrict__ W1,
                            float* __restrict__ H1) {
    const int wave = threadIdx.x >> 5;
    const int lane = threadIdx.x & 31;
    const int rb   = blockIdx.x * 8 + wave;      // 16-row block index (wave-uniform)
    if (rb >= N_NODES / 16) return;              // uniform exit: EXEC stays all-ones

    const int m    = lane & 15;                  // row within tile / B column
    const int half = lane >> 4;                  // 0: K{0,1}, 1: K{2,3}
    const int row  = rb * 16 + m;
    const float* xrow = X + (size_t)row * F_IN;

    v8f acc = {};
    #pragma unroll 8
    for (int k0 = 0; k0 < F_IN; k0 += 4) {
        const int ka = k0 + half * 2;
        // A fragment: 16x4 f32, lane holds X[row][ka], X[row][ka+1] (NT stream)
        v2f a = __builtin_nontemporal_load((const v2f*)(xrow + ka));
        a = dropout_pair(a, (unsigned)(row * F_IN + ka), 0x1234567u);
        // B fragment: 4x16 f32, lane holds W1[ka][m], W1[ka+1][m] (L2-hot, 32 KB)
        v2f b;
        b.x = W1[ka * HIDDEN + m];
        b.y = W1[(ka + 1) * HIDDEN + m];
        acc = __builtin_amdgcn_wmma_f32_16x16x4_f32(
                  false, a, false, b, (short)0, acc, false, false);
    }
    // C layout: VGPR r -> row (rb*16 + half*8 + r), col m
    float* out = H1 + (size_t)(rb * 16 + half * 8) * HIDDEN + m;
    #pragma unroll
    for (int r = 0; r < 8; ++r)
        out[(size_t)r * HIDDEN] = acc[r];
}

// ---------------------------------------------------------------------------
// COO SpMM with hardware f32 atomics: Hout[d] += w_e * Hin[s]  (L2 resident)
// One thread per (edge, channel).
// ---------------------------------------------------------------------------
template <int LOGC>
__global__ __launch_bounds__(256)
void gcn_spmm_atomic(const float* __restrict__ Hin,
                     const int*   __restrict__ esrc,
                     const int*   __restrict__ edst,
                     const float* __restrict__ ew,
                     float* __restrict__ Hout) {
    const unsigned gid   = blockIdx.x * 256u + threadIdx.x;
    const unsigned total = ((unsigned)N_EDGES) << LOGC;
    if (gid >= total) return;
    const unsigned e = gid >> LOGC;
    const unsigned c = gid & ((1u << LOGC) - 1u);
    const int   s = esrc[e];
    const int   d = edst[e];
    const float w = ew[e];
    const float v = Hin[((size_t)s << LOGC) + c];
    unsafeAtomicAdd(Hout + (((size_t)d << LOGC) + c), w * v);
}

// ---------------------------------------------------------------------------
// Layer 2 GEMM: H2 = dropout(relu(H1s)) @ W2   [N,16]@[16,32] -> [N,32]
// One wave = one 16-row tile, two 16x16 C tiles, 4 WMMA k-steps each.
// ---------------------------------------------------------------------------
__global__ __launch_bounds__(256)
void gcn_gemm2_relu_dropout_wmma(const float* __restrict__ H1s,
                                 const float* __restrict__ W2,
                                 float* __restrict__ H2) {
    const int wave = threadIdx.x >> 5;
    const int lane = threadIdx.x & 31;
    const int rb   = blockIdx.x * 8 + wave;
    if (rb >= N_NODES / 16) return;              // wave-uniform exit

    const int m    = lane & 15;
    const int half = lane >> 4;
    const int row  = rb * 16 + m;
    const float* hrow = H1s + (size_t)row * HIDDEN;

    v8f acc0 = {}, acc1 = {};
    #pragma unroll
    for (int k0 = 0; k0 < HIDDEN; k0 += 4) {
        const int ka = k0 + half * 2;
        v2f a;
        a.x = fmaxf(hrow[ka], 0.0f);
        a.y = fmaxf(hrow[ka + 1], 0.0f);
        a = dropout_pair(a, (unsigned)(row * HIDDEN + ka), 0x89ABCDEu);
        v2f b0, b1;
        b0.x = W2[ka * N_CLASSES + m];
        b0.y = W2[(ka + 1) * N_CLASSES + m];
        b1.x = W2[ka * N_CLASSES + 16 + m];
        b1.y = W2[(ka + 1) * N_CLASSES + 16 + m];
        acc0 = __builtin_amdgcn_wmma_f32_16x16x4_f32(
                   false, a, false, b0, (short)0, acc0, false, false);
        acc1 = __builtin_amdgcn_wmma_f32_16x16x4_f32(
                   false, a, false, b1, (short)0, acc1, false, false);
    }
    float* out = H2 + (size_t)(rb * 16 + half * 8) * N_CLASSES + m;
    #pragma unroll
    for (int r = 0; r < 8; ++r) {
        out[(size_t)r * N_CLASSES]      = acc0[r];
        out[(size_t)r * N_CLASSES + 16] = acc1[r];
    }
}

// ---------------------------------------------------------------------------
// Row softmax over 32 classes: one wave32 per row, shuffle reductions.
// ---------------------------------------------------------------------------
__global__ __launch_bounds__(256)
void gcn_softmax32(const float* __restrict__ Hin, float* __restrict__ Out) {
    const int wave = threadIdx.x >> 5;
    const int lane = threadIdx.x & 31;
    const int row  = blockIdx.x * 8 + wave;
    if (row >= N_NODES) return;
    float v  = Hin[(size_t)row * N_CLASSES + lane];
    float mx = v;
    #pragma unroll
    for (int off = 16; off; off >>= 1) mx = fmaxf(mx, __shfl_xor(mx, off));
    float ex  = __expf(v - mx);
    float sum = ex;
    #pragma unroll
    for (int off = 16; off; off >>= 1) sum += __shfl_xor(sum, off);
    Out[(size_t)row * N_CLASSES + lane] = ex / sum;
}

// ---------------------------------------------------------------------------
extern "C" void kernel_launch(void* const* d_in, const int* in_sizes, int n_in,
                              void* d_out, int out_size, void* d_ws, size_t ws_size,
                              hipStream_t stream) {
    const float* X    = (const float*)d_in[0];
    const int*   esrc = (const int*)  d_in[1];
    const int*   edst = (const int*)  d_in[2];
    const float* ew   = (const float*)d_in[3];
    const float* W1   = (const float*)d_in[4];
    const float* W2   = (const float*)d_in[5];
    float* out = (float*)d_out;

    // Workspace layout (25.6 MB total, H2s overlays dead H1/H1s):
    //   [0        .. 6.4MB )  H1   then later H2s (first half)
    //   [6.4MB    .. 12.8MB)  H1s  then later H2s (second half)
    //   [12.8MB   .. 25.6MB)  H2
    char* ws = (char*)d_ws;
    const size_t szH1 = (size_t)N_NODES * HIDDEN * sizeof(float);      // 6.4 MB
    float* H1  = (float*)(ws);
    float* H1s = (float*)(ws + szH1);
    float* H2  = (float*)(ws + 2 * szH1);
    float* H2s = (float*)(ws);  // reuses H1+H1s region after they are dead

    const int row_blocks = N_NODES / 16;                 // 6250 (exact)
    const int gemm_grid  = (row_blocks + 7) / 8;         // 782

    // 1) zero SpMM-1 accumulator
    hipMemsetAsync(H1s, 0, szH1, stream);
    // 2) H1 = dropout(X) @ W1
    gcn_gemm1_dropout_wmma<<<gemm_grid, 256, 0, stream>>>(X, W1, H1);
    // 3) H1s = A @ H1 (atomic scatter)
    {
        const unsigned total = (unsigned)N_EDGES * HIDDEN;     // 51.2M
        gcn_spmm_atomic<4><<<(total + 255) / 256, 256, 0, stream>>>(H1, esrc, edst, ew, H1s);
    }
    // 4) H2 = dropout(relu(H1s)) @ W2
    gcn_gemm2_relu_dropout_wmma<<<gemm_grid, 256, 0, stream>>>(H1s, W2, H2);
    // 5) zero SpMM-2 accumulator (H1/H1s now dead)
    hipMemsetAsync(H2s, 0, (size_t)N_NODES * N_CLASSES * sizeof(float), stream);
    // 6) H2s = A @ H2
    {
        const unsigned total = (unsigned)N_EDGES * N_CLASSES;  // 102.4M
        gcn_spmm_atomic<5><<<(total + 255) / 256, 256, 0, stream>>>(H2, esrc, edst, ew, H2s);
    }
    // 7) softmax rows -> d_out
    gcn_softmax32<<<(N_NODES + 7) / 8, 256, 0, stream>>>(H2s, out);
}